// Deepmd_angular_62328565399854
// MI455X (gfx1250) — compile-verified
//
#include <hip/hip_runtime.h>
#include <math.h>

// Problem constants from the reference
#define B_    8
#define N_    4096
#define M_    96
#define OUTW  384          // 3 * N_ANGULAR
#define RC_F  6.0f
#define PI_OVER_RC 0.52359877559829887308f   // pi / 6

constexpr int THREADS        = 256;           // 8 waves (wave32)
constexpr int WAVES          = THREADS / 32;  // 8
constexpr int ROWS_PER_BLOCK = 32;            // 4 rows per wave
constexpr int ROWS_PER_WAVE  = ROWS_PER_BLOCK / WAVES;

typedef unsigned long long u64;

// 128-bit vector pointee types with explicit address spaces (match the
// async-load builtin's parameter types).
typedef int v4i __attribute__((vector_size(16)));
typedef __attribute__((address_space(1))) v4i gv4i;   // global (AS1)
typedef __attribute__((address_space(3))) v4i lv4i;   // LDS (AS3)
typedef __attribute__((address_space(3))) void lvoid;

__global__ __launch_bounds__(THREADS)
void deepmd_angular_kernel(const float* __restrict__ pos,
                           const float* __restrict__ cell,
                           const float* __restrict__ off,
                           const float* __restrict__ msk,
                           const int*   __restrict__ nbr,
                           float*       __restrict__ out)
{
    __shared__ float posLDS[N_ * 3];                  // 48 KB: batch position table
    __shared__ __align__(16) u64 skey[WAVES][M_];     // 6 KB: per-wave packed sort keys

    const int blocksPerBatch = N_ / ROWS_PER_BLOCK;          // 128
    const int b       = blockIdx.x / blocksPerBatch;
    const int rowBase = (blockIdx.x % blocksPerBatch) * ROWS_PER_BLOCK;
    const int tid  = threadIdx.x;
    const int lane = tid & 31;
    const int wave = tid >> 5;

    // ---- CDNA5 async DMA: stage positions[b] (48 KB) into LDS -------------
    const float* posB = pos + (size_t)b * (N_ * 3);
    for (int c = tid; c < (N_ * 3) / 4; c += THREADS) {
        const float* g = posB + c * 4;
        float*       l = &posLDS[c * 4];
#if __has_builtin(__builtin_amdgcn_global_load_async_to_lds_b128)
        __builtin_amdgcn_global_load_async_to_lds_b128(
            (gv4i*)g, (lv4i*)l, 0, 0);
#else
        unsigned lds_off = (unsigned)(__SIZE_TYPE__)(lvoid*)l;
        unsigned g_off   = (unsigned)(c * 16);
        asm volatile("global_load_async_to_lds_b128 %0, %1, %2"
                     :: "v"(lds_off), "v"(g_off), "s"(posB) : "memory");
#endif
    }
#if __has_builtin(__builtin_amdgcn_s_wait_asynccnt)
    __builtin_amdgcn_s_wait_asynccnt(0);
#else
    asm volatile("s_wait_asynccnt 0" ::: "memory");
#endif
    __syncthreads();

    // cell[b] is a 3x3 row-major matrix, uniform per block -> SGPRs
    const float* cb = cell + b * 9;
    const float c00 = cb[0], c01 = cb[1], c02 = cb[2];
    const float c10 = cb[3], c11 = cb[4], c12 = cb[5];
    const float c20 = cb[6], c21 = cb[7], c22 = cb[8];

    for (int i = 0; i < ROWS_PER_WAVE; ++i) {
        const int n = rowBase + wave * ROWS_PER_WAVE + i;
        const size_t rowIdx = (size_t)b * N_ + n;
        const int*   nb = nbr + rowIdx * M_;
        const float* ob = off + rowIdx * (M_ * 3);
        const float* mb = msk + rowIdx * M_;

        const float pix = posLDS[n * 3 + 0];
        const float piy = posLDS[n * 3 + 1];
        const float piz = posLDS[n * 3 + 2];

        u64 K[3];
        float wx[3], wy[3], wz[3];

        // each lane owns neighbors m = lane, lane+32, lane+64
        #pragma unroll
        for (int t = 0; t < 3; ++t) {
            const int m = lane + 32 * t;
            const int j = nb[m];
            const float ox = ob[m * 3 + 0];
            const float oy = ob[m * 3 + 1];
            const float oz = ob[m * 3 + 2];
            const float mk = mb[m];
            const float pjx = posLDS[j * 3 + 0];
            const float pjy = posLDS[j * 3 + 1];
            const float pjz = posLDS[j * 3 + 2];
            // dis_vec = pos_j - pos_i + offsets @ cell   (einsum 'bnmi,bij->bnmj')
            const float dx = pjx - pix + ox * c00 + oy * c10 + oz * c20;
            const float dy = pjy - piy + ox * c01 + oy * c11 + oz * c21;
            const float dz = pjz - piz + ox * c02 + oy * c12 + oz * c22;
            const float d  = sqrtf(dx * dx + dy * dy + dz * dz + 1e-12f);
            float ct = 0.0f;
            if (mk != 0.0f && d < RC_F)
                ct = 0.5f * (cosf(PI_OVER_RC * d) + 1.0f) / d;
            wx[t] = ct * dx; wy[t] = ct * dy; wz[t] = ct * dz;
            // Packed stable key: cut >= 0, so float bits are order-monotonic.
            // Descending cut, ascending m on ties  <=>  strict descending u64.
            K[t] = ((u64)__float_as_uint(ct) << 32) | (u64)(0xFFFFFFFFu - (unsigned)m);
            skey[wave][m] = K[t];   // wave-private scratch
        }
        // wave-internal LDS RAW (cross-lane): same-wave DS ops are in order;
        // the wait + memory clobber keeps the compiler from reordering.
        asm volatile("s_wait_dscnt 0" ::: "memory");

        // rank = #{ j : K_j > K_m }  -> position in stable descending sort.
        // Read two keys per ds_load_b128 broadcast (uniform address).
        int r0 = 0, r1 = 0, r2 = 0;
        const ulonglong2* kp = (const ulonglong2*)&skey[wave][0];
        #pragma unroll 4
        for (int jj = 0; jj < M_ / 2; ++jj) {
            const ulonglong2 kk = kp[jj];
            r0 += (kk.x > K[0]) + (kk.y > K[0]);
            r1 += (kk.x > K[1]) + (kk.y > K[1]);
            r2 += (kk.x > K[2]) + (kk.y > K[2]);
        }

        float* orow = out + rowIdx * OUTW;
        orow[r0 * 3 + 0] = wx[0]; orow[r0 * 3 + 1] = wy[0]; orow[r0 * 3 + 2] = wz[0];
        orow[r1 * 3 + 0] = wx[1]; orow[r1 * 3 + 1] = wy[1]; orow[r1 * 3 + 2] = wz[1];
        orow[r2 * 3 + 0] = wx[2]; orow[r2 * 3 + 1] = wy[2]; orow[r2 * 3 + 2] = wz[2];

        // zero the padded tail columns [288, 384): 96 floats = 24 lanes * float4
        if (lane < 24) {
            float4 z = make_float4(0.f, 0.f, 0.f, 0.f);
            *(float4*)&orow[288 + lane * 4] = z;
        }
    }
}

extern "C" void kernel_launch(void* const* d_in, const int* in_sizes, int n_in,
                              void* d_out, int out_size, void* d_ws, size_t ws_size,
                              hipStream_t stream) {
    // setup_inputs() order: positions, cell, offsets, mask, neighbors
    const float* pos  = (const float*)d_in[0];
    const float* cell = (const float*)d_in[1];
    const float* off  = (const float*)d_in[2];
    const float* msk  = (const float*)d_in[3];
    const int*   nbr  = (const int*)d_in[4];
    float* out = (float*)d_out;

    dim3 grid(B_ * (N_ / ROWS_PER_BLOCK));   // 8 * 128 = 1024 blocks
    deepmd_angular_kernel<<<grid, THREADS, 0, stream>>>(pos, cell, off, msk, nbr, out);
}